// Predictor_69655779606914
// MI455X (gfx1250) — compile-verified
//
#include <hip/hip_runtime.h>
#include <math.h>

// ---------------------------------------------------------------------------
// Quadrotor MPC rollout for MI455X (gfx1250).
// Latency-bound sequential scan -> single wave32.
//  * fp32 WMMA (V_WMMA_F32_16X16X4_F32) for all MLP GEMVs.
//  * Weights pre-swizzled once into per-lane float2 B-fragments in LDS:
//    one unconditional ds_load_b64 per WMMA operand, no exec masking.
//  * A-fragment broadcast trick: all 16 A-rows = activation chunk, so every
//    D row (and both wave halves) holds the correct result -> branch-free
//    full-wave result store.
//  * Two interleaved accumulator chains per layer to hide WMMA RAW latency.
//  * Vehicle state + PID state live in per-lane redundant registers.
// ---------------------------------------------------------------------------

typedef __attribute__((ext_vector_type(2))) float v2f;
typedef __attribute__((ext_vector_type(8))) float v8f;

namespace {
constexpr int   N_PRED  = 300;
constexpr float TS_K    = 0.01f;
constexpr float M_Q     = 1.2f;
constexpr float GRAV    = 9.81f;
constexpr float L_ARM   = 0.16f;
constexpr float KTH     = 1.076e-05f;
constexpr float KTO     = 1.632e-07f;
constexpr float IXX     = 0.0123f;
constexpr float IYY     = 0.0123f;
constexpr float IZZ     = 0.0224f;
constexpr float TAU_M   = 0.015f;
constexpr float KP      = 2.0f;
constexpr float KI      = 0.5f;
constexpr float KD      = 0.1f;
constexpr float RADIUS  = 0.5f;

// fragment-buffer layout (units of v2f)
constexpr int FR_L0 = 0;                  // 2 tiles * 2 kc * 32 = 128
constexpr int FR_L1 = FR_L0 + 128;        // 2 tiles * 5 kc * 32 = 320
constexpr int FR_L2 = FR_L1 + 320;
constexpr int FR_L3 = FR_L2 + 320;
constexpr int FR_L4 = FR_L3 + 320;        // 1 tile  * 5 kc * 32 = 160
constexpr int FR_TOT = FR_L4 + 160;       // 1248 v2f = 9984 B
}

__device__ __forceinline__ float clip3(float v) { return fminf(fmaxf(v, -3.f), 3.f); }
__device__ __forceinline__ float clipw(float v) { return fminf(fmaxf(v, 100.f), 900.f); }

// One-time: swizzle W[FI][FO] (row-major, as in reference) into per-lane
// WMMA B fragments. frag[(t*NK+kc)*32 + lane] = {W[k0][col], W[k0+1][col]}
// with k0 = 4*kc + 2*(lane>>4), col = t*16 + (lane&15); zero-padded.
template <int FI, int FO>
__device__ __forceinline__ void build_frags(const float* __restrict__ gW,
                                            v2f* __restrict__ frag, int lane)
{
  constexpr int NK = (FI + 3) / 4;
  constexpr int NT = (FO + 15) / 16;
  const int n    = lane & 15;
  const int half = lane >> 4;
#pragma unroll
  for (int t = 0; t < NT; ++t) {
#pragma unroll
    for (int kc = 0; kc < NK; ++kc) {
      const int col = t * 16 + n;
      const int k0  = 4 * kc + 2 * half;
      v2f f;
      f[0] = (col < FO && (k0     < FI)) ? gW[(k0    ) * FO + col] : 0.f;
      f[1] = (col < FO && (k0 + 1 < FI)) ? gW[(k0 + 1) * FO + col] : 0.f;
      frag[(t * NK + kc) * 32 + lane] = f;
    }
  }
}

// h_out = act(h_in @ W + b).  hin: valid in [0,FI); hout: written in
// [0,32) (cols >= FO get exact zeros) plus shadow copy in [32,64).
template <int FI, int FO, bool RELU>
__device__ __forceinline__ void layer_wmma(const float* __restrict__ hin,
                                           float* __restrict__ hout,
                                           const v2f* __restrict__ frag,
                                           float bias0, float bias1, int lane)
{
  constexpr int NK = (FI + 3) / 4;
  constexpr int NT = (FO + 15) / 16;
  const int n    = lane & 15;
  const int half = lane >> 4;

  // A fragments: every lane loads the same K-chunk for its half-wave, so all
  // 16 A rows replicate the activation vector -> every D row is the result.
  v2f a[NK];
#pragma unroll
  for (int kc = 0; kc < NK; ++kc)
    a[kc] = *(const v2f*)(hin + 4 * kc + 2 * half);

  v2f b0[NK];
#pragma unroll
  for (int kc = 0; kc < NK; ++kc) b0[kc] = frag[kc * 32 + lane];
  v2f b1[NK];
  if constexpr (NT > 1) {
#pragma unroll
    for (int kc = 0; kc < NK; ++kc) b1[kc] = frag[(NK + kc) * 32 + lane];
  }

  v8f acc0 = {0.f, 0.f, 0.f, 0.f, 0.f, 0.f, 0.f, 0.f};
  v8f acc1 = {0.f, 0.f, 0.f, 0.f, 0.f, 0.f, 0.f, 0.f};
#pragma unroll
  for (int kc = 0; kc < NK; ++kc) {
    acc0 = __builtin_amdgcn_wmma_f32_16x16x4_f32(false, a[kc], false, b0[kc],
                                                 (short)0, acc0, false, false);
    if constexpr (NT > 1)
      acc1 = __builtin_amdgcn_wmma_f32_16x16x4_f32(false, a[kc], false, b1[kc],
                                                   (short)0, acc1, false, false);
  }

  float v0 = acc0[0] + bias0;
  if (RELU) v0 = fmaxf(v0, 0.f);
  hout[half * 32 + n] = v0;                 // upper half -> shadow region
  if constexpr (NT > 1) {
    float v1 = acc1[0] + bias1;
    if (RELU) v1 = fmaxf(v1, 0.f);
    hout[half * 32 + 16 + n] = v1;
  }
}

__global__ __launch_bounds__(32, 1)
void quad_rollout_kernel(
    const float* __restrict__ gx0,  const float* __restrict__ gpid,
    const float* __restrict__ gr,   const float* __restrict__ gcyl,
    const float* __restrict__ gdts,
    const float* __restrict__ gW0, const float* __restrict__ gb0,
    const float* __restrict__ gW1, const float* __restrict__ gb1,
    const float* __restrict__ gW2, const float* __restrict__ gb2,
    const float* __restrict__ gW3, const float* __restrict__ gb3,
    const float* __restrict__ gW4, const float* __restrict__ gb4,
    float* __restrict__ Xout, float* __restrict__ Uout, int* __restrict__ violOut)
{
  __shared__ v2f sFrag[FR_TOT];
  __shared__ __align__(16) float hA[64];
  __shared__ __align__(16) float hB[64];

  const int lane = threadIdx.x;
  const int n    = lane & 15;

  // ---- one-time weight swizzle into LDS fragments ----
  build_frags<8, 20>(gW0, sFrag + FR_L0, lane);
  build_frags<20, 20>(gW1, sFrag + FR_L1, lane);
  build_frags<20, 20>(gW2, sFrag + FR_L2, lane);
  build_frags<20, 20>(gW3, sFrag + FR_L3, lane);
  build_frags<20, 3>(gW4, sFrag + FR_L4, lane);

  // ---- biases in per-lane registers (identical across wave halves) ----
  const float bz = 0.f;
  float b0t0 = gb0[n], b0t1 = (16 + n < 20) ? gb0[16 + n] : bz;
  float b1t0 = gb1[n], b1t1 = (16 + n < 20) ? gb1[16 + n] : bz;
  float b2t0 = gb2[n], b2t1 = (16 + n < 20) ? gb2[16 + n] : bz;
  float b3t0 = gb3[n], b3t1 = (16 + n < 20) ? gb3[16 + n] : bz;
  float b4t0 = (n < 3) ? gb4[n] : bz;

  // ---- vehicle state: per-lane redundant registers ----
  float st[17];
#pragma unroll
  for (int i = 0; i < 17; ++i) st[i] = gx0[i];
  if (lane == 0) {
#pragma unroll
    for (int i = 0; i < 17; ++i) Xout[i] = st[i];
  }

  float integ0 = gpid[0], integ1 = gpid[1], integ2 = gpid[2];
  float pe0 = gpid[3], pe1 = gpid[4], pe2 = gpid[5];
  const float w_hover   = sqrtf(M_Q * GRAV / (4.f * KTH));
  const float hover_thr = M_Q * GRAV;

  float dvx = st[0] - 1.f, dvy = st[1] - 1.f;
  int viol = (dvx * dvx + dvy * dvy >= RADIUS * RADIUS) ? 1 : 0;

  __syncthreads();   // fragments visible before first layer

  for (int k = 0; k < N_PRED; ++k) {
    const float* rk = gr + k * 17;
    const float ck0 = gcyl[2 * k], ck1 = gcyl[2 * k + 1];
    const float dt  = gdts[k];
    if (k + 1 < N_PRED) __builtin_prefetch(gr + (k + 1) * 17, 0, 3);

    // ---- observation: SEL = [0,7,1,8,2,9], clip to [-3,3] ----
    float xr0 = clip3(st[0]), xr1 = clip3(st[7]), xr2 = clip3(st[1]);
    float xr3 = clip3(st[8]), xr4 = clip3(st[2]), xr5 = clip3(st[9]);
    float rr0 = clip3(rk[0]), rr1 = clip3(rk[7]), rr2 = clip3(rk[1]);
    float rr3 = clip3(rk[8]), rr4 = clip3(rk[2]), rr5 = clip3(rk[9]);
    float dx = xr0 - ck0, dy = xr2 - ck1;
    float dist  = sqrtf(dx * dx + dy * dy);
    float c_pos = dist - RADIUS;
    float c_vel = (xr1 * dx + xr3 * dy) / dist;
    if (lane == 0) {
      hA[0] = rr0 - xr0; hA[1] = rr1 - xr1; hA[2] = rr2 - xr2;
      hA[3] = rr3 - xr3; hA[4] = rr4 - xr4; hA[5] = rr5 - xr5;
      hA[6] = c_pos;     hA[7] = c_vel;
    }
    __syncthreads();

    // ---- MLP: 8->20->20->20->20->3, all via fp32 WMMA ----
    layer_wmma<8, 20, true >(hA, hB, sFrag + FR_L0, b0t0, b0t1, lane); __syncthreads();
    layer_wmma<20, 20, true>(hB, hA, sFrag + FR_L1, b1t0, b1t1, lane); __syncthreads();
    layer_wmma<20, 20, true>(hA, hB, sFrag + FR_L2, b2t0, b2t1, lane); __syncthreads();
    layer_wmma<20, 20, true>(hB, hA, sFrag + FR_L3, b3t0, b3t1, lane); __syncthreads();
    layer_wmma<20, 3, false>(hA, hB, sFrag + FR_L4, b4t0, bz,   lane); __syncthreads();

    const float u0 = hB[0], u1 = hB[1], u2 = hB[2] - hover_thr;

    // ---- PID (redundant on all lanes; identical results) ----
    float e0 = u0 - st[7], e1 = u1 - st[8], e2 = u2 - st[9];
    integ0 += e0 * TS_K; integ1 += e1 * TS_K; integ2 += e2 * TS_K;
    float d0 = (e0 - pe0) / TS_K, d1 = (e1 - pe1) / TS_K, d2 = (e2 - pe2) / TS_K;
    pe0 = e0; pe1 = e1; pe2 = e2;
    float a0 = KP * e0 + KI * integ0 + KD * d0;
    float a1 = KP * e1 + KI * integ1 + KD * d1;
    float a2 = KP * e2 + KI * integ2 + KD * d2;
    float wc0 = clipw(w_hover + 50.f * a0 + 50.f * a1 - 20.f * a2);
    float wc1 = clipw(w_hover - 50.f * a0 + 50.f * a1 + 20.f * a2);
    float wc2 = clipw(w_hover - 50.f * a0 - 50.f * a1 - 20.f * a2);
    float wc3 = clipw(w_hover + 50.f * a0 - 50.f * a1 + 20.f * a2);

    // ---- state_dot (redundant on all lanes, all in registers) ----
    float q0 = st[3], q1 = st[4], q2 = st[5], q3 = st[6];
    float om0 = st[10], om1 = st[11], om2 = st[12];
    float w0 = st[13], w1 = st[14], w2 = st[15], w3 = st[16];
    float w20 = w0 * w0, w21 = w1 * w1, w22 = w2 * w2, w23 = w3 * w3;
    float thrust = KTH * (w20 + w21 + w22 + w23);
    float zb0 = 2.f * (q1 * q3 + q0 * q2);
    float zb1 = 2.f * (q2 * q3 - q0 * q1);
    float zb2 = q0 * q0 - q1 * q1 - q2 * q2 + q3 * q3;
    float xd[17];
    xd[0] = st[7]; xd[1] = st[8]; xd[2] = st[9];
    xd[3] = 0.5f * (-q1 * om0 - q2 * om1 - q3 * om2);
    xd[4] = 0.5f * ( q0 * om0 + q2 * om2 - q3 * om1);
    xd[5] = 0.5f * ( q0 * om1 - q1 * om2 + q3 * om0);
    xd[6] = 0.5f * ( q0 * om2 + q1 * om1 - q2 * om0);
    xd[7] = zb0 * thrust / M_Q;
    xd[8] = zb1 * thrust / M_Q;
    xd[9] = -GRAV + zb2 * thrust / M_Q;
    float tau0 = L_ARM * KTH * (w23 - w21);
    float tau1 = L_ARM * KTH * (w20 - w22);
    float tau2 = KTO * (w20 - w21 + w22 - w23);
    float Ib0 = om0 * IXX, Ib1 = om1 * IYY, Ib2 = om2 * IZZ;
    xd[10] = (tau0 - (om1 * Ib2 - om2 * Ib1)) / IXX;
    xd[11] = (tau1 - (om2 * Ib0 - om0 * Ib2)) / IYY;
    xd[12] = (tau2 - (om0 * Ib1 - om1 * Ib0)) / IZZ;
    xd[13] = (wc0 - w0) / TAU_M;
    xd[14] = (wc1 - w1) / TAU_M;
    xd[15] = (wc2 - w2) / TAU_M;
    xd[16] = (wc3 - w3) / TAU_M;

#pragma unroll
    for (int i = 0; i < 17; ++i) st[i] = st[i] + dt * xd[i];
    float qs = sqrtf(st[3] * st[3] + st[4] * st[4] + st[5] * st[5] + st[6] * st[6]);
    st[3] /= qs; st[4] /= qs; st[5] /= qs; st[6] /= qs;

    float px1 = st[0] - 1.f, py1 = st[1] - 1.f;
    viol |= (px1 * px1 + py1 * py1 >= RADIUS * RADIUS) ? 1 : 0;

    if (lane == 0) {
#pragma unroll
      for (int i = 0; i < 17; ++i) Xout[(k + 1) * 17 + i] = st[i];
      Uout[4 * k + 0] = wc0; Uout[4 * k + 1] = wc1;
      Uout[4 * k + 2] = wc2; Uout[4 * k + 3] = wc3;
    }
  }

  if (lane == 0) *violOut = viol;
}

extern "C" void kernel_launch(void* const* d_in, const int* in_sizes, int n_in,
                              void* d_out, int out_size, void* d_ws, size_t ws_size,
                              hipStream_t stream) {
  (void)in_sizes; (void)n_in; (void)out_size; (void)d_ws; (void)ws_size;
  const float* x0  = (const float*)d_in[0];
  const float* pid = (const float*)d_in[1];
  const float* r   = (const float*)d_in[2];
  const float* cyl = (const float*)d_in[3];
  const float* dts = (const float*)d_in[4];
  const float* W0  = (const float*)d_in[5];
  const float* b0  = (const float*)d_in[6];
  const float* W1  = (const float*)d_in[7];
  const float* b1  = (const float*)d_in[8];
  const float* W2  = (const float*)d_in[9];
  const float* b2  = (const float*)d_in[10];
  const float* W3  = (const float*)d_in[11];
  const float* b3  = (const float*)d_in[12];
  const float* W4  = (const float*)d_in[13];
  const float* b4  = (const float*)d_in[14];

  float* X = (float*)d_out;           // (301, 17) = 5117 floats
  float* U = X + 301 * 17;            // (300, 4)  = 1200 floats
  int*   V = (int*)(U + 300 * 4);     // (1, 1) int32

  quad_rollout_kernel<<<1, 32, 0, stream>>>(
      x0, pid, r, cyl, dts,
      W0, b0, W1, b1, W2, b2, W3, b3, W4, b4,
      X, U, V);
}